// BlockDiagonalLinear_39298950758697
// MI455X (gfx1250) — compile-verified
//
#include <hip/hip_runtime.h>

// BlockDiagonalLinear for MI455X (gfx1250, wave32).
// y[t, 16g+4j+o] = sum_i x[t, 16g+4j+i] * W[4g+j][o][i]
// Each wave computes one 16-token x 16-column output tile with four chained
// v_wmma_f32_16x16x4_f32 ops (K split over the four 4x4 diagonal blocks).
// Memory-bound kernel: coalesced b128 streaming loads, nontemporal hints,
// LDS-staged transpose for the A fragment (bank-conflict-free, padded rows).

typedef __attribute__((ext_vector_type(2))) float v2f;
typedef __attribute__((ext_vector_type(4))) float v4f;
typedef __attribute__((ext_vector_type(8))) float v8f;

#define DIM   4096
#define CGRP  16                  // columns per WMMA tile (4 blocks of 4)
#define NCG   (DIM / CGRP)        // 256 column groups
#define WAVES 8
#define THREADS (WAVES * 32)
#define LDS_STRIDE 20             // floats per padded row: 80B (16B aligned, 16 distinct banks)

__global__ __launch_bounds__(THREADS)
void blockdiag_wmma_kernel(const float* __restrict__ x,
                           const float* __restrict__ W,
                           float* __restrict__ y,
                           int totalTiles) {
    __shared__ float lds_buf[WAVES][16][LDS_STRIDE];

    const int tid  = threadIdx.x;
    const int lane = tid & 31;
    const int wid  = tid >> 5;

    const int tileIdx = blockIdx.x * WAVES + wid;   // wave-uniform
    if (tileIdx >= totalTiles) return;              // whole wave exits together

    const int cg = tileIdx & (NCG - 1);             // column group (fast: block streams contiguous cols)
    const int tt = tileIdx >> 8;                    // token tile (16 tokens)

    const int half = lane >> 4;                     // 0: K 0..1 / M 0..7 half, 1: upper half
    const int n    = lane & 15;                     // N index (output col in group) / token row

    // ---- stage x tile (16 tokens x 16 cols) into LDS, coalesced b128 loads ----
    {
        const int r  = lane >> 1;                   // row 0..15
        const int ch = lane & 1;                    // 16B chunk 0/1 (second pair at +32B)
        const float* xp = x + (size_t)(tt * 16 + r) * DIM + cg * CGRP + ch * 4;
        v4f lo = __builtin_nontemporal_load((const v4f*)xp);
        v4f hi = __builtin_nontemporal_load((const v4f*)(xp + 8));
        float* srow = &lds_buf[wid][r][ch * 4];
        *(v4f*)(srow)     = lo;                     // ds_store_b128, 16B aligned
        *(v4f*)(srow + 8) = hi;
    }

    // ---- B fragments from W (tiny, L2-resident): lane holds B[K=v+2*half, N=n] ----
    // B_j[i, co] = (co>>2 == j) ? W[4*cg + j][co&3][i] : 0
    v2f bfrag[4];
    {
        const int o     = n & 3;
        const int jmine = n >> 2;
        const float* wp = W + (((size_t)cg * 4 + jmine) * 16 + o * 4 + half * 2);
        v2f wv;
        wv.x = wp[0];
        wv.y = wp[1];
#pragma unroll
        for (int j = 0; j < 4; ++j) {
            v2f z; z.x = 0.0f; z.y = 0.0f;
            bfrag[j] = (j == jmine) ? wv : z;
        }
    }

    // wave-local LDS RAW: stores above feed the transposed loads below
    asm volatile("s_wait_dscnt 0" ::: "memory");

    // ---- A fragments (transposed read, conflict-free): A_j[M=n, K=v+2*half] ----
    v2f afrag[4];
    {
        const float* lrow = &lds_buf[wid][n][0];
#pragma unroll
        for (int j = 0; j < 4; ++j)
            afrag[j] = *(const v2f*)(lrow + j * 4 + half * 2);   // ds_load_b64
    }

    // ---- four chained f32 WMMAs accumulate the full 16x16 tile ----
    v8f acc = {};
#pragma unroll
    for (int j = 0; j < 4; ++j) {
        acc = __builtin_amdgcn_wmma_f32_16x16x4_f32(
            /*neg_a=*/false, afrag[j],
            /*neg_b=*/false, bfrag[j],
            /*c_mod=*/(short)0, acc,
            /*reuse_a=*/false, /*reuse_b=*/false);
    }

    // ---- store D: element v -> y[row0 + v + 8*half, cg*16 + n] (coalesced b32) ----
    {
        float* yp = y + (size_t)(tt * 16 + half * 8) * DIM + cg * CGRP + n;
#pragma unroll
        for (int v = 0; v < 8; ++v) {
            __builtin_nontemporal_store(acc[v], yp);
            yp += DIM;
        }
    }
}

extern "C" void kernel_launch(void* const* d_in, const int* in_sizes, int n_in,
                              void* d_out, int out_size, void* d_ws, size_t ws_size,
                              hipStream_t stream) {
    const float* x = (const float*)d_in[0];   // [8, 2048, 4096] fp32
    const float* W = (const float*)d_in[1];   // [1024, 4, 4] fp32
    float* y = (float*)d_out;                 // [8, 2048, 4096] fp32

    const int tokens = in_sizes[0] / DIM;     // 16384
    const int ntt = tokens / 16;              // 1024 token tiles
    const int totalTiles = ntt * NCG;         // 262144 tiles, one per wave
    const int blocks = (totalTiles + WAVES - 1) / WAVES;

    blockdiag_wmma_kernel<<<blocks, THREADS, 0, stream>>>(x, W, y, totalTiles);
}